// NSGP_46961172415119
// MI455X (gfx1250) — compile-verified
//
#include <hip/hip_runtime.h>

// NSGP log-marginal-likelihood for MI455X (gfx1250, wave32).
// Heavy linear algebra runs on the fp32 matrix pipe (V_WMMA_F32_16X16X4_F32).
// All GEMM-shaped kernels (Cholesky SYRK, TRSM GEMM, W^T W) stage tiles with
// the Tensor Data Mover (TENSOR_LOAD_TO_LDS + TENSORcnt); SYRK and TRSM GEMM use the TDM
// pad feature to produce the 65-float LDS row stride needed for conflict-free
// fragment reads. Panel factorization / triangular solves are scalar fp32.

typedef __attribute__((ext_vector_type(2))) float v2f;
typedef __attribute__((ext_vector_type(8))) float v8f;
typedef unsigned int u32x4 __attribute__((ext_vector_type(4)));
typedef int i32x8 __attribute__((ext_vector_type(8)));
typedef int i32x4 __attribute__((ext_vector_type(4)));

#define NN 3072
#define DD 4
#define MM 512
#define JITTER 0.05f

// ---------------------------------------------------------------------------
// TDM helpers
// ---------------------------------------------------------------------------

// Generic pointer into LDS: low 32 bits are the LDS byte offset (aperture
// mapping truncates the upper bits).
__device__ inline unsigned lds_off(const void* p) {
  return (unsigned)(unsigned long long)p;
}

// DMA one 64x64 fp32 tile from a row-major matrix (row stride `ld` elements)
// into LDS at byte offset `lds_byte`. If `pad` != 0, insert 1 DWORD of LDS
// padding after every 64 DWORDs (row stride becomes 65 floats).
// 2D descriptor: groups 2/3 zero.
__device__ inline void tdm_load_tile64(const float* gsrc, unsigned lds_byte,
                                       unsigned ld, int pad) {
  unsigned long long ga = (unsigned long long)gsrc;
  u32x4 g0;
  g0[0] = 1u;                                   // count=1, user mode
  g0[1] = lds_byte;                             // lds_addr
  g0[2] = (unsigned)(ga & 0xffffffffu);         // global_addr[31:0]
  g0[3] = (unsigned)((ga >> 32) & 0x01ffffffu)  // global_addr[56:32]
          | (2u << 30);                         // type = 2 ("image")
  unsigned flags = 2u << 16;                    // wg_mask=0, data_size=4B
  if (pad) flags |= (1u << 20)                  // pad_enable
               | (5u << 22);                    // pad_interval: 64 DWORDs
                                                // pad_amount=0: 1 DWORD
  i32x8 g1;
  g1[0] = (int)flags;
  g1[1] = (int)((ld & 0xffffu) << 16);          // tensor_dim0[15:0]
  g1[2] = (int)(((ld >> 16) & 0xffffu)          // tensor_dim0[31:16]
          | (64u << 16));                       // tensor_dim1[15:0] = 64
  g1[3] = (int)(64u << 16);                     // tensor_dim1 hi=0, tile_dim0=64
  g1[4] = (int)64u;                             // tile_dim1=64, tile_dim2=0
  g1[5] = (int)ld;                              // tensor_dim0_stride[31:0]
  g1[6] = 0;                                    // stride hi, dim1_stride lo
  g1[7] = 0;
  i32x4 z4 = {0, 0, 0, 0};
  i32x8 z8 = {0, 0, 0, 0, 0, 0, 0, 0};
  __builtin_amdgcn_tensor_load_to_lds(g0, g1, z4, z4, z8, 0);
}

// ---------------------------------------------------------------------------
// Elementwise builders
// ---------------------------------------------------------------------------

__global__ void build_Kbar(float* KB, const float* Xb, const float* ls_a,
                           const float* std_a, const float* noise_a, int d) {
  int idx = blockIdx.x * blockDim.x + threadIdx.x;
  if (idx >= MM * MM) return;
  int i = idx / MM, j = idx % MM;
  float ls = ls_a[d], sd = std_a[d], nz = noise_a[d];
  float dx = Xb[i * DD + d] - Xb[j * DD + d];
  float v = sd * sd * __expf(-0.5f * dx * dx / (ls * ls));
  if (i == j) v += nz * nz;
  KB[idx] = v;
}

// KT[j][i] = k(x_i, xbar_j)   (M x N, stored row-major with ld = N)
__global__ void build_KT(float* KT, const float* X, const float* Xb,
                         const float* ls_a, const float* std_a, int d) {
  size_t idx = (size_t)blockIdx.x * blockDim.x + threadIdx.x;
  if (idx >= (size_t)MM * NN) return;
  int j = (int)(idx / NN), i = (int)(idx % NN);
  float ls = ls_a[d], sd = std_a[d];
  float dx = X[i * DD + d] - Xb[j * DD + d];
  KT[idx] = sd * sd * __expf(-0.5f * dx * dx / (ls * ls));
}

__global__ void build_Kglobal(float* K, const float* X, const float* lmat,
                              const float* gstd, const float* gnoise) {
  size_t idx = (size_t)blockIdx.x * blockDim.x + threadIdx.x;
  if (idx >= (size_t)NN * NN) return;
  int i = (int)(idx / NN), j = (int)(idx % NN);
  float suffix = 1.0f, expo = 0.0f;
  for (int d = 0; d < DD; ++d) {
    float l1 = lmat[i * DD + d], l2 = lmat[j * DD + d];
    float lsq = l1 * l1 + l2 * l2;
    suffix *= __fsqrt_rn(2.0f * l1 * l2 / lsq);
    float dx = X[i * DD + d] - X[j * DD + d];
    expo += dx * dx / lsq;
  }
  float gv = gstd[0] * gstd[0];
  float v = gv * suffix * __expf(-expo);
  if (i == j) v += gnoise[0] * gnoise[0];
  K[idx] = v;
}

// ---------------------------------------------------------------------------
// Blocked Cholesky building blocks (NB = 64)
// ---------------------------------------------------------------------------

__global__ void potrf64(float* A, int n, int kb) {
  __shared__ float T[64][65];
  int t = threadIdx.x;  // 64 threads
  int b = kb * 64;
  for (int c = 0; c < 64; ++c) T[t][c] = A[(size_t)(b + t) * n + b + c];
  __syncthreads();
  for (int j = 0; j < 64; ++j) {
    if (t == j) T[j][j] = __fsqrt_rn(T[j][j]);
    __syncthreads();
    if (t > j) T[t][j] /= T[j][j];
    __syncthreads();
    if (t > j) {
      float ltj = T[t][j];
      for (int c = j + 1; c <= t; ++c) T[t][c] -= ltj * T[c][j];
    }
    __syncthreads();
  }
  for (int c = 0; c <= t; ++c) A[(size_t)(b + t) * n + b + c] = T[t][c];
}

// Solve X * L11^T = A21 for one 64-row tile below the diagonal block.
__global__ void trsm64(float* A, int n, int kb) {
  __shared__ float Lk[64][65];
  __shared__ float R[64][65];
  int k0 = kb * 64;
  int r0 = (kb + 1 + blockIdx.x) * 64;
  int t = threadIdx.x;  // 64 threads
  for (int c = 0; c < 64; ++c) {
    Lk[t][c] = A[(size_t)(k0 + t) * n + k0 + c];
    R[t][c]  = A[(size_t)(r0 + t) * n + k0 + c];
  }
  __syncthreads();
  for (int c = 0; c < 64; ++c) {
    float s = R[t][c];
    for (int u = 0; u < c; ++u) s -= Lk[c][u] * R[t][u];
    R[t][c] = s / Lk[c][c];
  }
  __syncthreads();
  for (int c = 0; c < 64; ++c) A[(size_t)(r0 + t) * n + k0 + c] = R[t][c];
}

// Trailing update: A22(bi,bj) -= P_bi * P_bj^T, P = L[:, kb-panel].
// TDM-staged tiles (padded to stride 65), WMMA f32.
__global__ void syrk64(float* A, int n, int kb) {
  int bi = blockIdx.x, bj = blockIdx.y;
  if (bj > bi) return;
  __shared__ float Pa[64][65];
  __shared__ float Pb[64][65];
  int k0 = kb * 64;
  int r0 = (kb + 1 + bi) * 64, c0 = (kb + 1 + bj) * 64;
  int wave = threadIdx.x >> 5, lane = threadIdx.x & 31;
  if (wave == 0) {  // TDM issue once per workgroup (EXEC ignored by TDM)
    tdm_load_tile64(A + (size_t)r0 * n + k0, lds_off(&Pa[0][0]), n, 1);
    tdm_load_tile64(A + (size_t)c0 * n + k0, lds_off(&Pb[0][0]), n, 1);
    __builtin_amdgcn_s_wait_tensorcnt(0);
  }
  __syncthreads();
  int wr = (wave & 1) * 32, wc = (wave >> 1) * 32;  // 4 waves -> 2x2 of 32x32
  int lrow = lane & 15, khalf = lane >> 4;
  v8f acc[2][2] = {};
  for (int k4 = 0; k4 < 16; ++k4) {
    int kk = k4 * 4 + khalf * 2;
    v2f af[2], bf[2];
    for (int s = 0; s < 2; ++s) {
      af[s][0] = Pa[wr + s * 16 + lrow][kk + 0];
      af[s][1] = Pa[wr + s * 16 + lrow][kk + 1];
      bf[s][0] = Pb[wc + s * 16 + lrow][kk + 0];
      bf[s][1] = Pb[wc + s * 16 + lrow][kk + 1];
    }
    for (int si = 0; si < 2; ++si)
      for (int sj = 0; sj < 2; ++sj)
        acc[si][sj] = __builtin_amdgcn_wmma_f32_16x16x4_f32(
            false, af[si], false, bf[sj], (short)0, acc[si][sj], false, false);
  }
  for (int si = 0; si < 2; ++si)
    for (int sj = 0; sj < 2; ++sj)
      for (int r = 0; r < 8; ++r) {
        int row = r0 + wr + si * 16 + r + ((lane >= 16) ? 8 : 0);
        int col = c0 + wc + sj * 16 + (lane & 15);
        A[(size_t)row * n + col] -= acc[si][sj][r];
      }
}

// ---------------------------------------------------------------------------
// Multi-RHS forward solve:  W = L^{-1} * B   (B = KT in place, ldb = N)
// ---------------------------------------------------------------------------

__global__ void fsolve_diag(const float* L, int nL, float* B, int ldb, int kb) {
  __shared__ float Lk[64][65];
  int t = threadIdx.x;  // 64 threads
  int k0 = kb * 64;
  for (int c = 0; c < 64; ++c) Lk[t][c] = L[(size_t)(k0 + t) * nL + k0 + c];
  __syncthreads();
  int col = blockIdx.x * 64 + t;
  float x[64];
  for (int r = 0; r < 64; ++r) {
    float s = B[(size_t)(k0 + r) * ldb + col];
    for (int u = 0; u < r; ++u) s -= Lk[r][u] * x[u];
    x[r] = s / Lk[r][r];
  }
  for (int r = 0; r < 64; ++r) B[(size_t)(k0 + r) * ldb + col] = x[r];
}

// B[rows below, cols] -= L[rows, kb-panel] * B[kb-panel, cols]
// TDM-staged tiles (padded), WMMA f32.
__global__ void fsolve_gemm(const float* L, int nL, float* B, int ldb, int kb) {
  __shared__ float Lt[64][65];
  __shared__ float Wt[64][65];
  int k0 = kb * 64;
  int r0 = k0 + 64 + blockIdx.y * 64;
  int c0 = blockIdx.x * 64;
  int wave = threadIdx.x >> 5, lane = threadIdx.x & 31;
  if (wave == 0) {
    tdm_load_tile64(L + (size_t)r0 * nL + k0, lds_off(&Lt[0][0]), nL, 1);
    tdm_load_tile64(B + (size_t)k0 * ldb + c0, lds_off(&Wt[0][0]), ldb, 1);
    __builtin_amdgcn_s_wait_tensorcnt(0);
  }
  __syncthreads();
  int wr = (wave & 1) * 32, wc = (wave >> 1) * 32;
  int lrow = lane & 15, khalf = lane >> 4;
  v8f acc[2][2] = {};
  for (int k4 = 0; k4 < 16; ++k4) {
    int kk = k4 * 4 + khalf * 2;
    v2f af[2], bf[2];
    for (int s = 0; s < 2; ++s) {
      af[s][0] = Lt[wr + s * 16 + lrow][kk + 0];
      af[s][1] = Lt[wr + s * 16 + lrow][kk + 1];
      bf[s][0] = Wt[kk + 0][wc + s * 16 + lrow];
      bf[s][1] = Wt[kk + 1][wc + s * 16 + lrow];
    }
    for (int si = 0; si < 2; ++si)
      for (int sj = 0; sj < 2; ++sj)
        acc[si][sj] = __builtin_amdgcn_wmma_f32_16x16x4_f32(
            false, af[si], false, bf[sj], (short)0, acc[si][sj], false, false);
  }
  for (int si = 0; si < 2; ++si)
    for (int sj = 0; sj < 2; ++sj)
      for (int r = 0; r < 8; ++r) {
        int row = r0 + wr + si * 16 + r + ((lane >= 16) ? 8 : 0);
        int col = c0 + wc + sj * 16 + (lane & 15);
        B[(size_t)row * ldb + col] -= acc[si][sj][r];
      }
}

// ---------------------------------------------------------------------------
// Fused kpost tile: KP = kss - W^T W + jitter*I
// WMMA f32 over K = 512 with TDM double-buffered tile staging.
// ---------------------------------------------------------------------------

__global__ void kpost_tile(float* KP, const float* W, const float* X,
                           const float* ls_a, const float* std_a, int d) {
  __shared__ float Wa[2][64][64];
  __shared__ float Wb[2][64][64];
  int i0 = blockIdx.x * 64, j0 = blockIdx.y * 64;
  int wave = threadIdx.x >> 5, lane = threadIdx.x & 31;
  int wr = (wave & 1) * 32, wc = (wave >> 1) * 32;
  int lrow = lane & 15, khalf = lane >> 4;
  v8f acc[2][2] = {};
  const int NCH = MM / 64;  // 8 K-chunks

  if (wave == 0) {  // one TDM issue per workgroup (EXEC ignored by TDM)
    tdm_load_tile64(W + (size_t)0 * NN + i0, lds_off(&Wa[0][0][0]), NN, 0);
    tdm_load_tile64(W + (size_t)0 * NN + j0, lds_off(&Wb[0][0][0]), NN, 0);
  }
  for (int c = 0; c < NCH; ++c) {
    int cur = c & 1;
    __syncthreads();  // everyone done computing on the buffer we refill next
    if (wave == 0) {
      if (c + 1 < NCH) {
        const float* base = W + (size_t)(c + 1) * 64 * NN;
        tdm_load_tile64(base + i0, lds_off(&Wa[cur ^ 1][0][0]), NN, 0);
        tdm_load_tile64(base + j0, lds_off(&Wb[cur ^ 1][0][0]), NN, 0);
        __builtin_amdgcn_s_wait_tensorcnt(2);  // chunk c landed; c+1 in flight
      } else {
        __builtin_amdgcn_s_wait_tensorcnt(0);
      }
    }
    __syncthreads();  // release compute on buffer `cur`
    for (int k4 = 0; k4 < 16; ++k4) {
      int kk = k4 * 4 + khalf * 2;
      v2f af[2], bf[2];
      for (int s = 0; s < 2; ++s) {
        af[s][0] = Wa[cur][kk + 0][wr + s * 16 + lrow];
        af[s][1] = Wa[cur][kk + 1][wr + s * 16 + lrow];
        bf[s][0] = Wb[cur][kk + 0][wc + s * 16 + lrow];
        bf[s][1] = Wb[cur][kk + 1][wc + s * 16 + lrow];
      }
      for (int si = 0; si < 2; ++si)
        for (int sj = 0; sj < 2; ++sj)
          acc[si][sj] = __builtin_amdgcn_wmma_f32_16x16x4_f32(
              false, af[si], false, bf[sj], (short)0, acc[si][sj], false,
              false);
    }
  }
  float ls = ls_a[d], sd = std_a[d];
  float inv2 = 0.5f / (ls * ls), s2 = sd * sd;
  for (int si = 0; si < 2; ++si)
    for (int sj = 0; sj < 2; ++sj)
      for (int r = 0; r < 8; ++r) {
        int row = i0 + wr + si * 16 + r + ((lane >= 16) ? 8 : 0);
        int col = j0 + wc + sj * 16 + (lane & 15);
        float dx = X[row * DD + d] - X[col * DD + d];
        float v = s2 * __expf(-dx * dx * inv2) - acc[si][sj][r];
        if (row == col) v += JITTER;
        KP[(size_t)row * NN + col] = v;
      }
}

// ---------------------------------------------------------------------------
// Small vector kernels, triangular solves (single RHS), reductions
// ---------------------------------------------------------------------------

__global__ void alpha_init(float* a, const float* local_ls, int d) {
  int i = blockIdx.x * blockDim.x + threadIdx.x;
  if (i < MM) a[i] = __logf(fabsf(local_ls[i * DD + d]));
}

__global__ void copy_vec(float* dst, const float* src, int n) {
  int i = blockIdx.x * blockDim.x + threadIdx.x;
  if (i < n) dst[i] = src[i];
}

__global__ void zero_acc(float* acc) {
  if (threadIdx.x < 4) acc[threadIdx.x] = 0.0f;
}

// l[i,d] = exp( sum_j KT[j][i] * alpha[j] )
__global__ void gemv_l(float* lmat, const float* KT, const float* a, int d) {
  int i = blockIdx.x * blockDim.x + threadIdx.x;
  if (i >= NN) return;
  float s = 0.0f;
  for (int j = 0; j < MM; ++j) s += KT[(size_t)j * NN + i] * a[j];
  lmat[i * DD + d] = __expf(s);
}

// Single-workgroup forward substitution: L x = b (x in/out)
__global__ void fsub(const float* L, int n, float* x) {
  __shared__ float xs[NN];
  __shared__ float red[256];
  for (int i = threadIdx.x; i < n; i += blockDim.x) xs[i] = x[i];
  __syncthreads();
  for (int i = 0; i < n; ++i) {
    float s = 0.0f;
    for (int j = threadIdx.x; j < i; j += blockDim.x)
      s += L[(size_t)i * n + j] * xs[j];
    red[threadIdx.x] = s;
    __syncthreads();
    for (int o = 128; o > 0; o >>= 1) {
      if ((int)threadIdx.x < o) red[threadIdx.x] += red[threadIdx.x + o];
      __syncthreads();
    }
    if (threadIdx.x == 0) xs[i] = (xs[i] - red[0]) / L[(size_t)i * n + i];
    __syncthreads();
  }
  for (int i = threadIdx.x; i < n; i += blockDim.x) x[i] = xs[i];
}

// Single-workgroup back substitution: L^T x = b (x in/out)
__global__ void bsub(const float* L, int n, float* x) {
  __shared__ float xs[NN];
  __shared__ float red[256];
  for (int i = threadIdx.x; i < n; i += blockDim.x) xs[i] = x[i];
  __syncthreads();
  for (int i = n - 1; i >= 0; --i) {
    float s = 0.0f;
    for (int j = i + 1 + threadIdx.x; j < n; j += blockDim.x)
      s += L[(size_t)j * n + i] * xs[j];
    red[threadIdx.x] = s;
    __syncthreads();
    for (int o = 128; o > 0; o >>= 1) {
      if ((int)threadIdx.x < o) red[threadIdx.x] += red[threadIdx.x + o];
      __syncthreads();
    }
    if (threadIdx.x == 0) xs[i] = (xs[i] - red[0]) / L[(size_t)i * n + i];
    __syncthreads();
  }
  for (int i = threadIdx.x; i < n; i += blockDim.x) x[i] = xs[i];
}

__global__ void logdiag_sum(const float* A, int n, float* acc) {
  __shared__ float red[256];
  float s = 0.0f;
  for (int i = blockIdx.x * blockDim.x + threadIdx.x; i < n;
       i += gridDim.x * blockDim.x)
    s += __logf(A[(size_t)i * n + i]);
  red[threadIdx.x] = s;
  __syncthreads();
  for (int o = 128; o > 0; o >>= 1) {
    if ((int)threadIdx.x < o) red[threadIdx.x] += red[threadIdx.x + o];
    __syncthreads();
  }
  if (threadIdx.x == 0) atomicAdd(acc, red[0]);
}

__global__ void dot_sum(const float* a, const float* b, int n, float* acc) {
  __shared__ float red[256];
  float s = 0.0f;
  for (int i = blockIdx.x * blockDim.x + threadIdx.x; i < n;
       i += gridDim.x * blockDim.x)
    s += a[i] * b[i];
  red[threadIdx.x] = s;
  __syncthreads();
  for (int o = 128; o > 0; o >>= 1) {
    if ((int)threadIdx.x < o) red[threadIdx.x] += red[threadIdx.x + o];
    __syncthreads();
  }
  if (threadIdx.x == 0) atomicAdd(acc, red[0]);
}

__global__ void finalize(float* out, const float* acc) {
  if (threadIdx.x == 0) {
    float log2pi = 1.8378770664093453f;  // log(2*pi)
    float A = 0.5f * (acc[2] + acc[1] + (float)NN * log2pi);
    float B = acc[0] + 0.5f * (float)(MM * DD) * log2pi;
    out[0] = A + B;
  }
}

// ---------------------------------------------------------------------------
// Host-side launch orchestration
// ---------------------------------------------------------------------------

static void chol_blocked(float* A, int n, hipStream_t stream) {
  int nb = n / 64;
  for (int kb = 0; kb < nb; ++kb) {
    potrf64<<<1, 64, 0, stream>>>(A, n, kb);
    int below = nb - 1 - kb;
    if (below > 0) {
      trsm64<<<below, 64, 0, stream>>>(A, n, kb);
      syrk64<<<dim3(below, below), 128, 0, stream>>>(A, n, kb);
    }
  }
}

extern "C" void kernel_launch(void* const* d_in, const int* in_sizes, int n_in,
                              void* d_out, int out_size, void* d_ws,
                              size_t ws_size, hipStream_t stream) {
  (void)in_sizes; (void)n_in; (void)out_size; (void)ws_size;
  const float* X      = (const float*)d_in[0];  // (N, 4)
  const float* y      = (const float*)d_in[1];  // (N, 1)
  const float* Xb     = (const float*)d_in[2];  // (M, 4)
  const float* gls    = (const float*)d_in[3];  // (4,)
  const float* gstd_l = (const float*)d_in[4];  // (4,)
  const float* gnz_l  = (const float*)d_in[5];  // (4,)
  const float* lls    = (const float*)d_in[6];  // (M, 4)
  const float* gstd   = (const float*)d_in[7];  // (1,)
  const float* gnz    = (const float*)d_in[8];  // (1,)
  float* out = (float*)d_out;

  // Workspace layout (floats): ~45.2 MB total
  float* ws    = (float*)d_ws;
  float* KP    = ws;                          // N*N
  float* KT    = KP + (size_t)NN * NN;        // M*N (kstar^T, then W in place)
  float* KB    = KT + (size_t)MM * NN;        // M*M
  float* avec  = KB + (size_t)MM * MM;        // max(N, M)
  float* lmat  = avec + NN;                   // N*4
  float* acc   = lmat + (size_t)NN * DD;      // 4 scalars

  zero_acc<<<1, 32, 0, stream>>>(acc);

  const int TPB = 256;
  for (int d = 0; d < DD; ++d) {
    // K_bar and its Cholesky (512)
    build_Kbar<<<(MM * MM + TPB - 1) / TPB, TPB, 0, stream>>>(KB, Xb, gls,
                                                              gstd_l, gnz_l, d);
    chol_blocked(KB, MM, stream);

    // alpha = K_bar^{-1} log|local_ls[:,d]|
    alpha_init<<<(MM + TPB - 1) / TPB, TPB, 0, stream>>>(avec, lls, d);
    fsub<<<1, 256, 0, stream>>>(KB, MM, avec);
    bsub<<<1, 256, 0, stream>>>(KB, MM, avec);

    // kstar^T (M x N), then l[:,d] = exp(kstar @ alpha)
    build_KT<<<(int)(((size_t)MM * NN + TPB - 1) / TPB), TPB, 0, stream>>>(
        KT, X, Xb, gls, gstd_l, d);
    gemv_l<<<(NN + TPB - 1) / TPB, TPB, 0, stream>>>(lmat, KT, avec, d);

    // W = L^{-1} kstar^T (in place in KT), blocked forward TRSM
    for (int kb = 0; kb < MM / 64; ++kb) {
      fsolve_diag<<<NN / 64, 64, 0, stream>>>(KB, MM, KT, NN, kb);
      int below = MM / 64 - 1 - kb;
      if (below > 0)
        fsolve_gemm<<<dim3(NN / 64, below), 128, 0, stream>>>(KB, MM, KT, NN,
                                                              kb);
    }

    // kpost = kss - W^T W + jitter*I, then Cholesky(3072) and B_d
    kpost_tile<<<dim3(NN / 64, NN / 64), 128, 0, stream>>>(KP, KT, X, gls,
                                                           gstd_l, d);
    chol_blocked(KP, NN, stream);
    logdiag_sum<<<24, 256, 0, stream>>>(KP, NN, &acc[0]);
  }

  // Global GP: K, Cholesky, cho_solve(y), assemble A
  build_Kglobal<<<(int)(((size_t)NN * NN + TPB - 1) / TPB), TPB, 0, stream>>>(
      KP, X, lmat, gstd, gnz);
  chol_blocked(KP, NN, stream);
  logdiag_sum<<<24, 256, 0, stream>>>(KP, NN, &acc[1]);

  copy_vec<<<(NN + TPB - 1) / TPB, TPB, 0, stream>>>(avec, y, NN);
  fsub<<<1, 256, 0, stream>>>(KP, NN, avec);
  bsub<<<1, 256, 0, stream>>>(KP, NN, avec);
  dot_sum<<<24, 256, 0, stream>>>(y, avec, NN, &acc[2]);

  finalize<<<1, 32, 0, stream>>>(out, acc);
}